// SO3Linear_13254269075601
// MI455X (gfx1250) — compile-verified
//
#include <hip/hip_runtime.h>

typedef float v2f __attribute__((ext_vector_type(2)));
typedef float v4f __attribute__((ext_vector_type(4)));
typedef float v8f __attribute__((ext_vector_type(8)));

#define C_DIM   64
#define M_OUT   16
#define M_IN    16
#define K4CAP   176   // K=156 for this config; cap keeps LDS under 64KB static limit

// d_ws (int) layout:
//  [0]            : K4 (K rounded up to multiple of 4)
//  [16   .. 1040) : k_m1   (m1 index per compact k)
//  [1040 .. 2064) : k_lind (weight index per compact k)
//  [2064 .. +nnz) : j_kid  (compact k id per COO entry)

__global__ void so3_setup(const int* __restrict__ M1, const int* __restrict__ lind,
                          int nnz, int NW, int* __restrict__ ws) {
    __shared__ int s_cidx[1024];
    const int tid = threadIdx.x;
    const int P = M_IN * NW;  // <= 1024
    for (int p = tid; p < 1024; p += blockDim.x) s_cidx[p] = 0;
    __syncthreads();
    for (int j = tid; j < nnz; j += blockDim.x)
        s_cidx[M1[j] * NW + lind[j]] = 1;
    __syncthreads();
    if (tid == 0) {
        int K = 0;
        for (int p = 0; p < P; ++p) {
            if (s_cidx[p]) {
                ws[16 + K]   = p / NW;   // m1
                ws[1040 + K] = p % NW;   // lind
                s_cidx[p] = K;
                ++K;
            } else {
                s_cidx[p] = -1;
            }
        }
        int K4 = (K + 3) & ~3;
        for (int k = K; k < K4; ++k) { ws[16 + k] = 0; ws[1040 + k] = 0; }
        ws[0] = K4;
    }
    __syncthreads();
    for (int j = tid; j < nnz; j += blockDim.x)
        ws[2064 + j] = s_cidx[M1[j] * NW + lind[j]];
}

__global__ __launch_bounds__(128) void so3_main(
    const float* __restrict__ feature,  // B x 16 x 64
    const float* __restrict__ edge,     // B x 16
    const float* __restrict__ weight,   // NW x 64
    const float* __restrict__ cg,       // nnz
    const int*   __restrict__ M2,       // nnz
    const int*   __restrict__ Mseg,     // nnz
    const int*   __restrict__ ws,
    float*       __restrict__ out,      // B x 16 x 64
    int nnz) {
    __shared__ __align__(16) float s_feat[M_IN * C_DIM];        // 4 KB
    __shared__ float s_ef[M_IN];
    __shared__ __align__(16) float s_A[M_OUT * K4CAP];          // 11 KB, A[M][k]
    __shared__ __align__(16) float s_FW[K4CAP * C_DIM];         // 44 KB, FW[k][c]

    const int tid  = threadIdx.x;
    const int wave = tid >> 5;
    const int lane = tid & 31;
    const int b    = blockIdx.x;

    const int  K4     = ws[0];          // uniform -> scalar load
    const int* k_m1   = ws + 16;
    const int* k_lind = ws + 1040;
    const int* j_kid  = ws + 2064;

    // Stage feature tile (1024 floats = 4 KB) into LDS with the gfx1250 async
    // global->LDS DMA (ASYNCcnt path): no VGPR round-trip, DS pipe stays free.
    // GV mode: VDST = per-lane LDS byte offset, VADDR = 64-bit global address.
    {
        unsigned long long gsrc =
            (unsigned long long)(feature + (size_t)b * (M_IN * C_DIM)) + (unsigned)tid * 16u;
        unsigned lds_off = (unsigned)(size_t)(void*)s_feat + (unsigned)tid * 16u;
        asm volatile("global_load_async_to_lds_b128 %0, %1, off"
                     :: "v"(lds_off), "v"(gsrc) : "memory");
        gsrc    += 128u * 16u;
        lds_off += 128u * 16u;
        asm volatile("global_load_async_to_lds_b128 %0, %1, off"
                     :: "v"(lds_off), "v"(gsrc) : "memory");
    }
    if (tid < M_IN) s_ef[tid] = edge[b * M_IN + tid];

    // Zero the A tile (padding columns stay zero -> no contribution).
    {
        v4f z = {0.f, 0.f, 0.f, 0.f};
        v4f* adst = (v4f*)s_A;
        for (int i = tid; i < (M_OUT * K4CAP) / 4; i += 128) adst[i] = z;
    }
    asm volatile("s_wait_asynccnt 0" ::: "memory");
    __syncthreads();

    // Build FW[k][c] = feat[m1_k][c] * w[lind_k][c]; wave-uniform k -> scalar idx loads.
    for (int k = wave; k < K4; k += 4) {
        const int m1 = k_m1[k];
        const int li = k_lind[k];
        const int c2 = lane * 2;
        v2f f = *(const v2f*)(s_feat + m1 * C_DIM + c2);
        v2f w = *(const v2f*)(weight + li * C_DIM + c2);
        v2f r; r.x = f.x * w.x; r.y = f.y * w.y;
        *(v2f*)(s_FW + k * C_DIM + c2) = r;
    }

    // Scatter A[Mseg_j][kid_j] += CG_j * ef[M2_j]  (ds_add_f32).
    for (int j = tid; j < nnz; j += 128) {
        float d = cg[j] * s_ef[M2[j]];
        atomicAdd(&s_A[Mseg[j] * K4 + j_kid[j]], d);
    }
    __syncthreads();

    // out[b][:, cbase:cbase+16] = A (16xK4) @ FW (K4x64) via chained f32 WMMA.
    const int cbase = wave * 16;
    const int half  = lane >> 4;   // 0: lanes 0-15, 1: lanes 16-31
    const int l15   = lane & 31 & 15;
    v8f acc = {0.f, 0.f, 0.f, 0.f, 0.f, 0.f, 0.f, 0.f};
    const int ktiles = K4 >> 2;
    for (int kt = 0; kt < ktiles; ++kt) {
        const int k0 = kt * 4 + half * 2;
        // A frag: VGPR0 = A[M][k0], VGPR1 = A[M][k0+1]  (K{0,2}/K{1,3} split at lane 16)
        v2f a = *(const v2f*)(s_A + l15 * K4 + k0);
        // B frag: mirrored layout, rows striped across lanes within a VGPR
        v2f bf;
        bf.x = s_FW[(k0    ) * C_DIM + cbase + l15];
        bf.y = s_FW[(k0 + 1) * C_DIM + cbase + l15];
        acc = __builtin_amdgcn_wmma_f32_16x16x4_f32(false, a, false, bf,
                                                    (short)0, acc, false, false);
    }

    // C/D layout: VGPR i -> M = i + 8*half, N = l15.
    float* op = out + (size_t)b * (M_OUT * C_DIM) + cbase + l15;
#pragma unroll
    for (int i = 0; i < 8; ++i)
        op[(i + half * 8) * C_DIM] = acc[i];
}

extern "C" void kernel_launch(void* const* d_in, const int* in_sizes, int n_in,
                              void* d_out, int out_size, void* d_ws, size_t ws_size,
                              hipStream_t stream) {
    const float* feature = (const float*)d_in[0];
    const float* edge    = (const float*)d_in[1];
    const float* weight  = (const float*)d_in[2];
    const float* cg      = (const float*)d_in[3];
    const int*   M1      = (const int*)d_in[4];
    const int*   M2      = (const int*)d_in[5];
    const int*   lind    = (const int*)d_in[6];
    const int*   Mseg    = (const int*)d_in[7];

    const int nnz = in_sizes[3];
    const int NW  = in_sizes[2] / C_DIM;          // num_weights (=34 here)
    const int B   = in_sizes[0] / (M_IN * C_DIM); // 4096

    int* ws = (int*)d_ws;
    so3_setup<<<1, 256, 0, stream>>>(M1, lind, nnz, NW, ws);
    so3_main<<<B, 128, 0, stream>>>(feature, edge, weight, cg, M2, Mseg, ws,
                                    (float*)d_out, nnz);
}